// ArchetipesNetwork_86603720557272
// MI455X (gfx1250) — compile-verified
//
#include <hip/hip_runtime.h>
#include <hip/hip_bf16.h>

// Problem constants (from reference)
#define T_STEPS 2048
#define NINP    128
#define NMOD    64
#define NHID    256
#define DT_C    0.01f

// padded LDS row stride for h2h (260 % 64 == 4 -> rotates banks per row)
#define HSTRIDE 260
// padded LDS row stride for x tile in the GEMM (132 % 64 == 4)
#define XSTRIDE 132

typedef float v2f __attribute__((ext_vector_type(2)));
typedef float v8f __attribute__((ext_vector_type(8)));
typedef int   v4i __attribute__((ext_vector_type(4)));

// ---------------------------------------------------------------------------
// CDNA5 async global->LDS copy (ASYNCcnt-tracked, bypasses VGPRs), with a
// register-path fallback if the builtin is unavailable on this toolchain.
// Probe-confirmed signature: (v4i AS1*, v4i AS3*, imm offset, imm cpol).
// ---------------------------------------------------------------------------
#if defined(__has_builtin)
#  if __has_builtin(__builtin_amdgcn_global_load_async_to_lds_b128)
#    define HAVE_ASYNC_LDS 1
#  endif
#endif
#ifndef HAVE_ASYNC_LDS
#  define HAVE_ASYNC_LDS 0
#endif

__device__ __forceinline__ void async_g2l_b128(const float* g, float* l) {
#if HAVE_ASYNC_LDS
  __builtin_amdgcn_global_load_async_to_lds_b128(
      (__attribute__((address_space(1))) v4i*)g,
      (__attribute__((address_space(3))) v4i*)l,
      0, 0);
#else
  float4 v = *(const float4*)g;
  *(float4*)l = v;
#endif
}

__device__ __forceinline__ void async_wait0() {
#if HAVE_ASYNC_LDS
#  if __has_builtin(__builtin_amdgcn_s_wait_asynccnt)
  __builtin_amdgcn_s_wait_asynccnt(0);
#  else
  asm volatile("s_wait_asynccnt 0" ::: "memory");
#  endif
#endif
}

// ---------------------------------------------------------------------------
// Init: zero grid-barrier counter, zero outs ping buffer (outs0 == 0 in the
// reference scan carry), copy initial_states into states_seq[0].
// ---------------------------------------------------------------------------
__global__ void __launch_bounds__(256)
init_kernel(const float* __restrict__ init_states,
            float* __restrict__ out_states0,   // d_out, first 64*2*256 floats
            float* __restrict__ outs_buf0,     // ping buffer (64*256)
            unsigned* __restrict__ counter) {
  int i = blockIdx.x * blockDim.x + threadIdx.x;
  if (i == 0) *counter = 0u;
  if (i < NMOD * 2 * NHID) out_states0[i] = init_states[i];
  if (i < NMOD * NHID)     outs_buf0[i]   = 0.0f;
}

// ---------------------------------------------------------------------------
// XP precompute: XP[t, nh] = sum_i x2h[nh, i] * x[t, i]
// fp32 GEMM (16384 x 2048 x 128) on V_WMMA_F32_16X16X4_F32.
//  - A (x2h K-panel, 16 rows x 128 K) stationary in 64 VGPRs per wave
//  - B (x tile, 16 t-rows x 128 K) staged once per block into padded LDS via
//    async load-to-LDS and shared by all 8 waves (identical t-range)
// ---------------------------------------------------------------------------
__global__ void __launch_bounds__(256)
xp_gemm_kernel(const float* __restrict__ x,     // (T, 128)
               const float* __restrict__ x2h,   // (16384, 128) rows = n*256+h
               float* __restrict__ xp) {        // (T, 16384)
  __shared__ float xtile[16 * XSTRIDE];          // 16 t-rows x 128 K, padded

  const int tid  = threadIdx.x;
  const int lane = tid & 31;
  const int wave = tid >> 5;
  const int nh0  = (blockIdx.x * 8 + wave) * 16;   // 1024 nh-tiles total
  const int m    = lane & 15;                      // row within tile (A/M, B/N)
  const int kh   = (lane >> 4) * 2;                // K sub-pair: 0 or 2

  // A: x2h tile rows nh0..nh0+15, all K=0..127 -> 32 x v2f (stationary)
  v2f a[32];
#pragma unroll
  for (int kk = 0; kk < 32; ++kk) {
    const float* p = x2h + (size_t)(nh0 + m) * NINP + kk * 4 + kh;
    a[kk][0] = p[0];
    a[kk][1] = p[1];
  }

#pragma unroll 1
  for (int j = 0; j < 8; ++j) {
    const int t0 = (blockIdx.y * 8 + j) * 16;      // 128 t-tiles total

    // stage x[t0 .. t0+16) x 128 into LDS: 512 float4s over 256 threads
#pragma unroll
    for (int e4 = tid; e4 < 512; e4 += 256) {
      const int trow = e4 >> 5;                    // 32 float4 per row
      const int c4   = (e4 & 31) * 4;
      async_g2l_b128(x + (size_t)(t0 + trow) * NINP + c4,
                     &xtile[trow * XSTRIDE + c4]);
    }
    async_wait0();
    __syncthreads();

    v8f c = {};                                    // fresh accumulator
#pragma unroll
    for (int kk = 0; kk < 32; ++kk) {
      const float* q = &xtile[m * XSTRIDE + kk * 4 + kh];
      v2f b;
      b[0] = q[0];
      b[1] = q[1];
      // 8-arg pattern: (neg_a, A, neg_b, B, c_mod, C, reuse_a, reuse_b)
      c = __builtin_amdgcn_wmma_f32_16x16x4_f32(
              false, a[kk], false, b, (short)0, c, false, false);
    }
    // D layout: VGPR g -> (M = g + 8*(lane>=16), N = lane&15)
#pragma unroll
    for (int g = 0; g < 8; ++g) {
      xp[(size_t)(t0 + m) * (NMOD * NHID) + nh0 + g + (lane >> 4) * 8] = c[g];
    }
    __syncthreads();   // xtile reuse fence before next j stages over it
  }
}

// ---------------------------------------------------------------------------
// Persistent recurrent kernel: 64 blocks (one per module) x 512 threads.
// Thread (r, half) with r = tid&255, half = tid>>8:
//   - holds Wm[r, half*128 .. +128) in 128 VGPRs (stationary all 2048 steps)
//   - h2h_n staged into LDS (stride-260 padded) via async load-to-LDS
//   - per step: s = C_row @ outs_prev; fb = Wm @ s; hc = H @ hy; xc from XP;
//     tanh/Euler update; hy_new -> double-buffered global outs; then a
//     device-wide atomic barrier (release/acquire, agent scope).
// ---------------------------------------------------------------------------
__global__ void __launch_bounds__(512)
recurrent_kernel(const float* __restrict__ x,          // (T, 128)
                 const float* __restrict__ init_states,// (64, 2, 256)
                 const float* __restrict__ wm,         // (256, 256)
                 const float* __restrict__ conn,       // (64, 64)
                 const float* __restrict__ x2h,        // (64, 256, 128)
                 const float* __restrict__ h2h,        // (64, 256, 256)
                 const float* __restrict__ bias,       // (64, 256)
                 const float* __restrict__ xp,         // (T, 16384) or unused
                 float* __restrict__ out_states,       // (T+1, 64, 2, 256)
                 float* __restrict__ out_fb,           // (T, 64, 256)
                 float* __restrict__ outs_buf,         // 2 x (64*256)
                 unsigned* __restrict__ counter,
                 int use_xp) {
  extern __shared__ float Hs[];          // 256 * HSTRIDE floats (~260 KB)
  __shared__ float s_sh[NHID];
  __shared__ float hy_sh[NHID];
  __shared__ float hz_sh[NHID];
  __shared__ float red[512];
  __shared__ float redW[512];
  __shared__ float crow[NMOD];
  __shared__ float xs[NINP];

  const int n    = blockIdx.x;
  const int tid  = threadIdx.x;
  const int r    = tid & 255;
  const int half = tid >> 8;
  const int o0   = half * 128;

  // --- stationary Wm half-row in VGPRs -------------------------------------
  float wrow[128];
  {
    const float* wp = wm + (size_t)r * NHID + o0;
#pragma unroll
    for (int j = 0; j < 128; ++j) wrow[j] = wp[j];
  }

  // --- stage h2h_n into padded LDS (async, 16384 float4s / 512 threads) ----
  {
    const float* hp = h2h + (size_t)n * NHID * NHID;
#pragma unroll 4
    for (int e4 = tid; e4 < (NHID * NHID) / 4; e4 += 512) {
      const int rr = e4 >> 6;                      // 64 float4 per row
      const int oo = (e4 & 63) * 4;
      async_g2l_b128(hp + (size_t)e4 * 4, &Hs[rr * HSTRIDE + oo]);
    }
    async_wait0();
  }
  if (tid < NMOD) crow[tid] = conn[n * NMOD + tid];
  if (tid < NHID) {
    hy_sh[tid] = init_states[(size_t)n * 2 * NHID + tid];
    hz_sh[tid] = init_states[(size_t)n * 2 * NHID + NHID + tid];
  }
  const float bias_r = bias[(size_t)n * NHID + r];
  __syncthreads();

  unsigned bar_target = 0;

  for (int t = 0; t < T_STEPS; ++t) {
    const float* prev = outs_buf + (size_t)(t & 1) * (NMOD * NHID);
    float*       next = outs_buf + (size_t)((t + 1) & 1) * (NMOD * NHID);

    if (!use_xp && tid < NINP) xs[tid] = x[(size_t)t * NINP + tid];

    // s[h] = sum_m C[n,m] * outs_prev[m,h]  (split m over halves)
    {
      float ps = 0.0f;
      const int m0 = half * 32;
#pragma unroll
      for (int mm = 0; mm < 32; ++mm)
        ps = fmaf(crow[m0 + mm], prev[(m0 + mm) * NHID + r], ps);
      red[tid] = ps;
    }
    __syncthreads();
    if (tid < NHID) s_sh[tid] = red[tid] + red[tid + 256];
    __syncthreads();

    // fb partial: Wm row from registers, s broadcast from LDS
    float accW = 0.0f;
#pragma unroll
    for (int j = 0; j < 128; ++j)
      accW = fmaf(wrow[j], s_sh[o0 + j], accW);

    // h2h partial: row slice from padded LDS, hy broadcast from LDS
    float accO = 0.0f;
    {
      const float* hr = &Hs[r * HSTRIDE + o0];
#pragma unroll 8
      for (int j = 0; j < 128; ++j)
        accO = fmaf(hr[j], hy_sh[o0 + j], accO);
    }

    // x2h fallback (only when XP buffer unavailable)
    if (!use_xp) {
      const float* xr = x2h + ((size_t)n * NHID + r) * NINP + half * 64;
#pragma unroll 8
      for (int j = 0; j < 64; ++j)
        accO = fmaf(xr[j], xs[half * 64 + j], accO);
    }

    redW[tid] = accW;
    red[tid]  = accO;
    __syncthreads();

    if (tid < NHID) {
      float fb  = redW[tid] + redW[tid + 256];
      float acc = red[tid] + red[tid + 256];
      float xc  = use_xp ? xp[(size_t)t * (NMOD * NHID) + n * NHID + tid] : 0.0f;
      float pre = acc + xc + bias_r + fb;
      float hy = hy_sh[tid], hz = hz_sh[tid];
      float hz_new = fmaf(DT_C, tanhf(pre) - hy - hz, hz);  // GAMMA=EPS=1
      float hy_new = fmaf(DT_C, hz_new, hy);
      hy_sh[tid] = hy_new;
      hz_sh[tid] = hz_new;

      const size_t so = ((size_t)(t + 1) * NMOD + n) * (2 * NHID);
      out_states[so + tid]        = hy_new;
      out_states[so + NHID + tid] = hz_new;
      out_fb[((size_t)t * NMOD + n) * NHID + tid] = fb;
      next[n * NHID + tid] = hy_new;
    }

    // ---- device-wide barrier (monotonic counter, sense-free) --------------
    bar_target += NMOD;
    __syncthreads();
    if (tid == 0) {
      __hip_atomic_fetch_add(counter, 1u, __ATOMIC_RELEASE,
                             __HIP_MEMORY_SCOPE_AGENT);
      while (__hip_atomic_load(counter, __ATOMIC_ACQUIRE,
                               __HIP_MEMORY_SCOPE_AGENT) < bar_target)
        __builtin_amdgcn_s_sleep(1);
    }
    __syncthreads();
  }
}

// ---------------------------------------------------------------------------
extern "C" void kernel_launch(void* const* d_in, const int* in_sizes, int n_in,
                              void* d_out, int out_size, void* d_ws, size_t ws_size,
                              hipStream_t stream) {
  const float* x    = (const float*)d_in[0];
  const float* ist  = (const float*)d_in[1];
  const float* wm   = (const float*)d_in[2];
  const float* conn = (const float*)d_in[3];
  const float* x2h  = (const float*)d_in[4];
  const float* h2h  = (const float*)d_in[5];
  const float* bias = (const float*)d_in[6];

  float* out        = (float*)d_out;
  float* out_states = out;                                           // (2049,64,2,256)
  float* out_fb     = out + (size_t)(T_STEPS + 1) * NMOD * 2 * NHID; // (2048,64,256)

  char* ws = (char*)d_ws;
  unsigned* counter = (unsigned*)ws;                 // 256 B slot
  float* outs_buf   = (float*)(ws + 256);            // 2 x 64*256 floats
  const size_t base     = 256 + 2 * (size_t)NMOD * NHID * sizeof(float);
  const size_t xp_bytes = (size_t)T_STEPS * NMOD * NHID * sizeof(float); // 134 MB
  const int use_xp = (ws_size >= base + xp_bytes) ? 1 : 0;
  float* xp = (float*)(ws + base);

  init_kernel<<<128, 256, 0, stream>>>(ist, out_states, outs_buf, counter);

  if (use_xp) {
    dim3 g(128, 16);   // 1024 nh-tiles x 128 t-tiles, 8 waves/block, 8 t-tiles/wave
    xp_gemm_kernel<<<g, 256, 0, stream>>>(x, x2h, xp);
  }

  const size_t lds_bytes = (size_t)NHID * HSTRIDE * sizeof(float); // ~260 KB
  (void)hipFuncSetAttribute(reinterpret_cast<const void*>(recurrent_kernel),
                            hipFuncAttributeMaxDynamicSharedMemorySize,
                            (int)lds_bytes);
  recurrent_kernel<<<NMOD, 512, lds_bytes, stream>>>(
      x, ist, wm, conn, x2h, h2h, bias, xp,
      out_states, out_fb, outs_buf, counter, use_xp);
}